// WKNNFingerprintModel_35742717837535
// MI455X (gfx1250) — compile-verified
//
#include <hip/hip_runtime.h>
#include <hip/hip_bf16.h>

typedef __attribute__((ext_vector_type(2))) float v2f;
typedef __attribute__((ext_vector_type(8))) float v8f;
typedef __attribute__((ext_vector_type(4))) unsigned int u32x4;
typedef __attribute__((ext_vector_type(8))) unsigned int u32x8;

#define F_DIM 128
#define BT 64                 // rows per workgroup
#define NT 32                 // cols per subtile
#define SUBTILES 64           // 32*64 = 2048 cols per chunk
#define CHUNK_COLS (NT * SUBTILES)
#define XS_STRIDE 132         // 528 B rows: 16B aligned, bank-stride 4 (conflict-free halves)
#define FS_STRIDE 132
#define DT_STRIDE 33
#define BIGF 3.402823466e38f

__device__ __forceinline__ void topk_insert(float d, int g, float td[4], int ti[4]) {
    if (d < td[3]) {
        if (d < td[2]) {
            td[3] = td[2]; ti[3] = ti[2];
            if (d < td[1]) {
                td[2] = td[1]; ti[2] = ti[1];
                if (d < td[0]) { td[1] = td[0]; ti[1] = ti[0]; td[0] = d; ti[0] = g; }
                else           { td[1] = d;     ti[1] = g; }
            } else { td[2] = d; ti[2] = g; }
        } else { td[3] = d; ti[3] = g; }
    }
}

// Issue a TDM load of a (32 rows x 128 floats) fingerprint tile into LDS,
// padding 4 DWORDs after every 128 DWORDs -> LDS row stride 132 floats.
// D# layout per CDNA5 ISA (group0 128b, group1 256b).
__device__ __forceinline__ void tdm_load_tile(const float* gsrc, unsigned lds_off,
                                              unsigned rows_remaining) {
    unsigned long long ga = (unsigned long long)(uintptr_t)gsrc;
    u32x4 g0;
    g0.x = 0x1u;                                    // count=1, user mode
    g0.y = lds_off;                                 // lds_addr (bytes)
    g0.z = (unsigned)(ga & 0xFFFFFFFFu);            // global_addr[31:0]
    g0.w = (unsigned)((ga >> 32) & 0x01FFFFFFu)     // global_addr[56:32]
         | (2u << 30);                              // type=2 ("image")
    u32x8 g1;
    g1[0] = (2u << 16)                              // data_size = 4 bytes
          | (1u << 20)                              // pad_enable
          | (6u << 22)                              // pad_interval: 128 DWORDs
          | (3u << 25);                             // pad_amount: 4 DWORDs
    g1[1] = (unsigned)F_DIM << 16;                  // tensor_dim0[15:0] = 128
    g1[2] = (rows_remaining & 0xFFFFu) << 16;       // tensor_dim1[15:0]
    g1[3] = ((rows_remaining >> 16) & 0xFFFFu)      // tensor_dim1[31:16]
          | ((unsigned)F_DIM << 16);                // tile_dim0 = 128
    g1[4] = (unsigned)NT;                           // tile_dim1 = 32, tile_dim2 = 0
    g1[5] = (unsigned)F_DIM;                        // tensor_dim0_stride = 128 elements
    g1[6] = 0u;
    g1[7] = 0u;
    asm volatile("tensor_load_to_lds %0, %1" :: "s"(g0), "s"(g1) : "memory");
}

// Kernel 0: f2[n] = sum_k fingerprints[n][k]^2. One wave per row.
__global__ __launch_bounds__(256)
void wknn_f2(const float* __restrict__ Fp, float* __restrict__ f2, int N) {
    int wave = threadIdx.x >> 5, lane = threadIdx.x & 31;
    int n = blockIdx.x * 8 + wave;
    if (n >= N) return;
    float4 v = *(const float4*)(Fp + (size_t)n * F_DIM + lane * 4);
    float s = v.x * v.x + v.y * v.y + v.z * v.z + v.w * v.w;
    #pragma unroll
    for (int off = 16; off > 0; off >>= 1) s += __shfl_xor(s, off);
    if (lane == 0) f2[n] = s;
}

// Kernel 1: fused (f2 - 2*x.f) GEMM via fp32 WMMA + per-chunk top-4 per row.
// TDM double-buffers fingerprint tiles into LDS while waves run WMMAs.
// gridDim.x = N-chunks, gridDim.y = B/64. block = 256 threads (8 waves).
__global__ __launch_bounds__(256)
void wknn_gemm_topk(const float* __restrict__ X,   // (B, 128)
                    const float* __restrict__ Fp,  // (N, 128)
                    const float* __restrict__ f2g, // (N,)
                    float* __restrict__ wsD,       // (B, nchunks, 4) score = f2-2dot
                    int*   __restrict__ wsI,       // (B, nchunks, 4) idx
                    int B, int N, int nchunks)
{
    __shared__ __attribute__((aligned(16))) float Xs[BT * XS_STRIDE];
    __shared__ __attribute__((aligned(16))) float Fs[2][NT * FS_STRIDE];
    __shared__ __attribute__((aligned(16))) float Dt[BT * DT_STRIDE]; // 2112 floats
    __shared__ float f2s[2][NT];

    const int tid  = threadIdx.x;
    const int lane = tid & 31;
    const int w    = tid >> 5;
    const int half = lane >> 4;
    const int l16  = lane & 15;
    const int m0   = (w & 3) * 16;
    const int n0   = (w >> 2) * 16;

    const int rowBase = blockIdx.y * BT;
    const int colBase = blockIdx.x * CHUNK_COLS;

    const unsigned ldsFs0 = (unsigned)(uintptr_t)&Fs[0][0];
    const unsigned ldsFs1 = (unsigned)(uintptr_t)&Fs[1][0];

    // ---- stage X tile (64 x 128) into LDS, coalesced float4 ----
    {
        const float4* Xg = (const float4*)(X + (size_t)rowBase * F_DIM);
        for (int i = tid; i < BT * (F_DIM / 4); i += 256) {
            int r = i >> 5, kq = i & 31;
            float4 v = Xg[r * (F_DIM / 4) + kq];
            *(float4*)&Xs[r * XS_STRIDE + kq * 4] = v;
        }
    }
    // ---- TDM prologue: DMA first fingerprint tile into Fs[0] ----
    if (w == 0) {
        if (colBase < N)
            tdm_load_tile(Fp + (size_t)colBase * F_DIM, ldsFs0, (unsigned)(N - colBase));
        __builtin_amdgcn_s_wait_tensorcnt(0);
    }
    __syncthreads();

    float td[4] = {BIGF, BIGF, BIGF, BIGF};
    int   ti[4] = {0, 0, 0, 0};

    for (int s = 0; s < SUBTILES; ++s) {
        const int cur  = s & 1;
        const int colS = colBase + s * NT;

        // f2 for the current tile (published by B1, consumed in epilogue)
        if (tid < NT) {
            int n = colS + tid;
            f2s[cur][tid] = (n < N) ? f2g[n] : 0.f;
        }
        // TDM: DMA next tile into the other buffer while we compute
        if (w == 0 && s + 1 < SUBTILES) {
            int colNext = colS + NT;
            if (colNext < N)
                tdm_load_tile(Fp + (size_t)colNext * F_DIM,
                              cur ? ldsFs0 : ldsFs1, (unsigned)(N - colNext));
        }

        // ---- 16x16 dot-product tile via V_WMMA_F32_16X16X4_F32 ----
        v8f acc = {};
        const float* xrow = &Xs[(m0 + l16) * XS_STRIDE + 2 * half];
        const float* frow = &Fs[cur][(n0 + l16) * FS_STRIDE + 2 * half];
        #pragma unroll
        for (int kk = 0; kk < F_DIM; kk += 4) {
            v2f a = *(const v2f*)(xrow + kk);
            v2f b = *(const v2f*)(frow + kk);
            acc = __builtin_amdgcn_wmma_f32_16x16x4_f32(
                false, a, false, b, (short)0, acc, false, false);
        }
        __syncthreads();   // B1: f2s[cur] visible; prior scan of Dt finished

        // ---- epilogue: score = |f|^2 - 2*dot (row-constant |x|^2 deferred) ----
        const float f2v = f2s[cur][n0 + l16];
        #pragma unroll
        for (int v = 0; v < 8; ++v) {
            int mrow = m0 + v + 8 * half;
            Dt[mrow * DT_STRIDE + (n0 + l16)] = f2v - 2.0f * acc[v];
        }
        __syncthreads();   // B2: Dt visible

        // ---- running top-4: row = tid&63 scans 8 cols of its phase ----
        {
            int row = tid & 63, phase = tid >> 6;
            #pragma unroll
            for (int j = 0; j < 8; ++j) {
                int col = phase * 8 + j;
                int g = colS + col;
                float dv = (g < N) ? Dt[row * DT_STRIDE + col] : BIGF;
                topk_insert(dv, g, td, ti);
            }
        }
        if (w == 0) __builtin_amdgcn_s_wait_tensorcnt(0);
        __syncthreads();   // B3: next Fs buffer ready; Fs[cur]/Dt reads done
    }

    // ---- merge 4 phases per row -> chunk-level top-4, write to workspace ----
    float* mD = Dt;                    // 1024 floats
    int*   mI = (int*)(Dt + BT * 16);  // 1024 ints (Dt has 2112 slots)
    {
        int row = tid & 63, phase = tid >> 6;
        int base = (row * 4 + phase) * 4;
        #pragma unroll
        for (int q = 0; q < 4; ++q) { mD[base + q] = td[q]; mI[base + q] = ti[q]; }
    }
    __syncthreads();
    if (tid < BT) {
        float bd[4] = {BIGF, BIGF, BIGF, BIGF};
        int   bi[4] = {0, 0, 0, 0};
        int base = tid * 16;
        for (int c = 0; c < 16; ++c) topk_insert(mD[base + c], mI[base + c], bd, bi);
        int grow = rowBase + tid;
        long off = ((long)grow * nchunks + blockIdx.x) * 4;
        #pragma unroll
        for (int q = 0; q < 4; ++q) { wsD[off + q] = bd[q]; wsI[off + q] = bi[q]; }
    }
}

// Kernel 2: final top-4 merge + inverse-distance weighting of positions.
__global__ __launch_bounds__(256)
void wknn_reduce(const float* __restrict__ wsD, const int* __restrict__ wsI,
                 const float* __restrict__ X, const float* __restrict__ Pos,
                 float* __restrict__ out, int B, int nchunks)
{
    int r = blockIdx.x * 256 + threadIdx.x;
    if (r >= B) return;
    float bd[4] = {BIGF, BIGF, BIGF, BIGF};
    int   bi[4] = {0, 0, 0, 0};
    long base = (long)r * nchunks * 4;
    int total = nchunks * 4;
    for (int c = 0; c < total; ++c) topk_insert(wsD[base + c], wsI[base + c], bd, bi);

    // |x|^2 for this row (scores stored were f2 - 2*dot)
    float x2 = 0.f;
    const float4* xr = (const float4*)(X + (size_t)r * F_DIM);
    #pragma unroll 8
    for (int i = 0; i < F_DIM / 4; ++i) {
        float4 v = xr[i];
        x2 += v.x * v.x + v.y * v.y + v.z * v.z + v.w * v.w;
    }

    float wsum = 0.f, px = 0.f, py = 0.f;
    #pragma unroll
    for (int q = 0; q < 4; ++q) {
        float d2 = x2 + bd[q];
        float d = sqrtf(fmaxf(d2, 0.f) + 1e-12f);
        float t = d + 1e-6f;
        float wv = 1.0f / (t * t);               // P = 2
        wsum += wv;
        px += Pos[(size_t)bi[q] * 2 + 0] * wv;
        py += Pos[(size_t)bi[q] * 2 + 1] * wv;
    }
    float inv = 1.0f / (wsum + 1e-12f);
    out[r * 2 + 0] = px * inv;
    out[r * 2 + 1] = py * inv;
}

extern "C" void kernel_launch(void* const* d_in, const int* in_sizes, int n_in,
                              void* d_out, int out_size, void* d_ws, size_t ws_size,
                              hipStream_t stream) {
    const float* X   = (const float*)d_in[0];  // (B,128)
    const float* Fp  = (const float*)d_in[1];  // (N,128)
    const float* Pos = (const float*)d_in[2];  // (N,2)
    float* out = (float*)d_out;

    int B = in_sizes[0] / F_DIM;
    int N = in_sizes[2] / 2;
    int nchunks = (N + CHUNK_COLS - 1) / CHUNK_COLS;

    size_t candCount = (size_t)B * nchunks * 4;
    float* wsD = (float*)d_ws;
    int*   wsI = (int*)((char*)d_ws + candCount * sizeof(float));
    float* f2  = (float*)((char*)d_ws + candCount * (sizeof(float) + sizeof(int)));

    wknn_f2<<<(N + 7) / 8, 256, 0, stream>>>(Fp, f2, N);
    dim3 g1(nchunks, B / BT);
    wknn_gemm_topk<<<g1, 256, 0, stream>>>(X, Fp, f2, wsD, wsI, B, N, nchunks);
    wknn_reduce<<<(B + 255) / 256, 256, 0, stream>>>(wsD, wsI, X, Pos, out, B, nchunks);
}